// soft_cldice_49108656062751
// MI455X (gfx1250) — compile-verified
//
#include <hip/hip_runtime.h>

// ---------------- problem geometry ----------------
#define HH 512
#define WW 512
#define NPLANES 64                 // 32 batches * 2 channels
#define PLANE_ELEMS (HH*WW)        // 262144

// ---------------- stencil tiling ------------------
#define TILE 64
#define XDIM 68                    // TILE + 2*2 halo
#define XS   69                    // LDS stride for x tile
#define RMW  66                    // row-min width   (XDIM - 2)
#define RS   67                    // row-min / mt LDS stride
#define MDIM 66                    // minpool tile    (TILE + 2)
#define RXS  65                    // row-max LDS stride (reuses tmp)

typedef __attribute__((ext_vector_type(2))) float v2f;
typedef __attribute__((ext_vector_type(8))) float v8f;

__device__ __forceinline__ unsigned lds_byte_off(const void* p) {
  // Generic LDS addresses carry the workgroup-relative LDS offset in the
  // low 32 bits -- exactly what the async
  // load's VDST (LDS address) operand wants.
  return (unsigned)(size_t)p;
}

// ---------------------------------------------------------------------------
// One soft_skeletonize iteration, fully fused with separable pooling:
//   m = minpool3(x) (clipped)   = colmin3(rowmin3(x))
//   contour = relu(maxpool3(m) - m),  maxpool3 = colmax3(rowmax3(m))
//   out = relu(x - contour)
// Input tile staged into LDS with CDNA5 async global->LDS copies (ASYNCcnt).
// ---------------------------------------------------------------------------
__global__ __launch_bounds__(256) void skel_iter(const float* __restrict__ src,
                                                 float* __restrict__ dst) {
  __shared__ float xt[XDIM * XS];     // 68x68 input tile (stride 69)
  __shared__ float tmp[XDIM * RS];    // row-min (68x66), later row-max (66x64)
  __shared__ float mt[MDIM * RS];     // 66x66 minpool tile

  const int tid   = threadIdx.x;
  const int x0    = blockIdx.x * TILE;
  const int y0    = blockIdx.y * TILE;
  const int plane = blockIdx.z;
  const float* sp = src + (size_t)plane * PLANE_ELEMS;
  float*       dp = dst + (size_t)plane * PLANE_ELEMS;
  const float INF = __builtin_inff();

  // ---- Phase A: stage (TILE+4)^2 region; +inf at out-of-plane halo -------
  for (int l = tid; l < XDIM * XDIM; l += 256) {
    int ly = l / XDIM, lx = l - ly * XDIM;
    int gy = y0 - 2 + ly, gx = x0 - 2 + lx;
    if (gy >= 0 && gy < HH && gx >= 0 && gx < WW) {
      unsigned           ldst = lds_byte_off(&xt[ly * XS + lx]);
      unsigned long long gsrc = (unsigned long long)(sp + gy * WW + gx);
      asm volatile("global_load_async_to_lds_b32 %0, %1, off"
                   :: "v"(ldst), "v"(gsrc) : "memory");
    } else {
      xt[ly * XS + lx] = INF;   // clipped-window min semantics
    }
  }
  asm volatile("s_wait_asynccnt 0" ::: "memory");
  __syncthreads();

  // ---- Phase B1: horizontal min3 (68 rows x 66 cols) ----------------------
  for (int l = tid; l < XDIM * RMW; l += 256) {
    int y = l / RMW, i = l - y * RMW;
    const float* r = &xt[y * XS + i];
    tmp[y * RS + i] = fminf(fminf(r[0], r[1]), r[2]);
  }
  __syncthreads();

  // ---- Phase B2: vertical min3 -> mt; -inf where center out-of-plane -----
  for (int l = tid; l < MDIM * MDIM; l += 256) {
    int j = l / MDIM, i = l - j * MDIM;
    int py = y0 - 1 + j, px = x0 - 1 + i;
    float m;
    if (py < 0 || py >= HH || px < 0 || px >= WW) {
      m = -INF;                 // excluded from the following max-pool
    } else {
      m = fminf(fminf(tmp[j * RS + i], tmp[(j + 1) * RS + i]),
                tmp[(j + 2) * RS + i]);
    }
    mt[j * RS + i] = m;
  }
  __syncthreads();              // also makes tmp reusable below

  // ---- Phase C1: horizontal max3 of mt (66 rows x 64 cols), reuse tmp ----
  for (int l = tid; l < MDIM * TILE; l += 256) {
    int j = l / TILE, i = l - j * TILE;
    const float* r = &mt[j * RS + i];
    tmp[j * RXS + i] = fmaxf(fmaxf(r[0], r[1]), r[2]);
  }
  __syncthreads();

  // ---- Phase C2: vertical max3 + contour + update, 64x64 outputs ---------
  for (int q = tid; q < TILE * TILE; q += 256) {
    int oy = q / TILE, ox = q - oy * TILE;
    float maxv = fmaxf(fmaxf(tmp[oy * RXS + ox], tmp[(oy + 1) * RXS + ox]),
                       tmp[(oy + 2) * RXS + ox]);
    float c       = mt[(oy + 1) * RS + (ox + 1)];
    float contour = fmaxf(maxv - c, 0.0f);
    float xv      = xt[(oy + 2) * XS + (ox + 2)];
    dp[(y0 + oy) * WW + (x0 + ox)] = fmaxf(xv - contour, 0.0f);
  }
}

// ---------------------------------------------------------------------------
// WMMA reduction over the channel-1 planes (8,388,608 elems):
//   sums[0] += sum(skel * other),  sums[1] += sum(skel)
// A = ones(16x4)  =>  D = A*B + C accumulates column sums of B; the total of
// the 16x16 accumulator equals 16 * sum(B), so each v_wmma_f32_16x16x4_f32
// folds 64 f32 values per wave in full f32 precision.
// Launch: 2048 blocks x 256 threads = 16384 waves; 8 iters x 32 pairs/wave
// covers the domain exactly (no predication -> EXEC all-ones for WMMA).
// ---------------------------------------------------------------------------
__global__ __launch_bounds__(256) void reduce_wmma(const float* __restrict__ skel,
                                                   const float* __restrict__ other,
                                                   float* __restrict__ sums) {
  __shared__ float acc[2];
  const int tid = threadIdx.x;
  if (tid < 2) acc[tid] = 0.0f;
  __syncthreads();

  const unsigned lane = tid & 31;
  const unsigned wid  = blockIdx.x * 8u + (tid >> 5);   // wave id in grid

  v2f a; a.x = 1.0f; a.y = 1.0f;                         // ones A-matrix
  v8f cp = {};                                           // product accumulator
  v8f cs = {};                                           // plain-sum accumulator

#pragma unroll 1
  for (unsigned it = 0; it < 8; ++it) {
    unsigned j    = (it * 16384u + wid) * 32u + lane;    // pair index
    unsigned i    = 2u * j;                              // element index in domain
    unsigned n    = i >> 18;                             // batch
    unsigned memi = (n << 19) + (1u << 18) + (i & 262143u); // channel-1 plane
    float2 s = *(const float2*)(skel  + memi);
    float2 t = *(const float2*)(other + memi);
    v2f bp; bp.x = s.x * t.x; bp.y = s.y * t.y;
    v2f bs; bs.x = s.x;       bs.y = s.y;
    cp = __builtin_amdgcn_wmma_f32_16x16x4_f32(false, a, false, bp, (short)0, cp, false, false);
    cs = __builtin_amdgcn_wmma_f32_16x16x4_f32(false, a, false, bs, (short)0, cs, false, false);
  }

  float lp = 0.0f, ls = 0.0f;
#pragma unroll
  for (int k = 0; k < 8; ++k) { lp += cp[k]; ls += cs[k]; }
  atomicAdd(&acc[0], lp);
  atomicAdd(&acc[1], ls);
  __syncthreads();
  if (tid == 0) {
    atomicAdd(&sums[0], acc[0] * (1.0f / 16.0f));
    atomicAdd(&sums[1], acc[1] * (1.0f / 16.0f));
  }
}

__global__ void zero4(float* sums) {
  if (threadIdx.x < 4) sums[threadIdx.x] = 0.0f;
}

__global__ void finalize(const float* __restrict__ sums, float* __restrict__ out) {
  float s1 = sums[0], s2 = sums[1], s3 = sums[2], s4 = sums[3];
  float tprec = (s1 + 1.0f) / (s2 + 1.0f);
  float tsens = (s3 + 1.0f) / (s4 + 1.0f);
  out[0] = 1.0f - 2.0f * (tprec * tsens) / (tprec + tsens);
}

// ---------------------------------------------------------------------------
extern "C" void kernel_launch(void* const* d_in, const int* in_sizes, int n_in,
                              void* d_out, int out_size, void* d_ws, size_t ws_size,
                              hipStream_t stream) {
  (void)in_sizes; (void)n_in; (void)out_size; (void)ws_size;
  const float* y_pred = (const float*)d_in[0];
  const float* y_true = (const float*)d_in[1];
  float* out = (float*)d_out;

  char*  ws   = (char*)d_ws;
  const size_t BUF_BYTES = (size_t)NPLANES * PLANE_ELEMS * sizeof(float); // 64 MiB
  float* bufA = (float*)(ws);
  float* bufB = (float*)(ws + BUF_BYTES);
  float* sums = (float*)(ws + 2 * BUF_BYTES);   // 4 floats

  zero4<<<1, 4, 0, stream>>>(sums);

  dim3 grid(WW / TILE, HH / TILE, NPLANES);   // 8 x 8 x 64
  dim3 block(256);

  // skeletonize y_pred (10 iterations, ping-pong; both buffers stay in L2)
  const float* cur = y_pred;
  float* wa = bufA; float* wb = bufB;
  for (int i = 0; i < 10; ++i) {
    skel_iter<<<grid, block, 0, stream>>>(cur, wa);
    cur = wa; float* t = wa; wa = wb; wb = t;
  }
  reduce_wmma<<<2048, 256, 0, stream>>>(cur, y_true, sums + 0);  // S1, S2

  // skeletonize y_true
  cur = y_true; wa = bufA; wb = bufB;
  for (int i = 0; i < 10; ++i) {
    skel_iter<<<grid, block, 0, stream>>>(cur, wa);
    cur = wa; float* t = wa; wa = wb; wb = t;
  }
  reduce_wmma<<<2048, 256, 0, stream>>>(cur, y_pred, sums + 2);  // S3, S4

  finalize<<<1, 1, 0, stream>>>(sums, out);
}